// KMLayer_67516885893220
// MI455X (gfx1250) — compile-verified
//
#include <hip/hip_runtime.h>
#include <cmath>

typedef float v2f __attribute__((ext_vector_type(2)));
typedef float v8f __attribute__((ext_vector_type(8)));

constexpr int SPLITK = 8;   // K-slices per m-tile -> 8x more waves for latency hiding

// ---------------------------------------------------------------------------
// GroupNorm over c: groups of 2 channels x N elements, write y in [col][n]
// layout, col = b*16 + ch.  One block per (b, group).
// ---------------------------------------------------------------------------
__global__ void gn_kernel(const float* __restrict__ cin,
                          const float* __restrict__ w,
                          const float* __restrict__ bb,
                          float* __restrict__ y,
                          int N, int G, int C) {
    int b = blockIdx.x / G;
    int g = blockIdx.x % G;
    int chPer = C / G;                       // 2
    const float* base = cin + ((size_t)b * C + (size_t)g * chPer) * N;
    int tid = threadIdx.x;
    int total = chPer * N;
    float s = 0.f, ss = 0.f;
    for (int i = tid; i < total; i += 256) {
        float v = base[i];
        s += v; ss += v * v;
    }
    __shared__ float sh1[256], sh2[256];
    sh1[tid] = s; sh2[tid] = ss;
    __syncthreads();
    for (int off = 128; off > 0; off >>= 1) {
        if (tid < off) { sh1[tid] += sh1[tid + off]; sh2[tid] += sh2[tid + off]; }
        __syncthreads();
    }
    float mu   = sh1[0] / (float)total;
    float var  = sh2[0] / (float)total - mu * mu;
    float rstd = rsqrtf(var + 1e-5f);
    for (int i = tid; i < total; i += 256) {
        int chi = i / N;
        int n   = i - chi * N;
        int ch  = g * chPer + chi;
        float v = (base[i] - mu) * rstd * w[ch] + bb[ch];
        y[((size_t)(b * C + ch)) * N + n] = v;
    }
}

// ---------------------------------------------------------------------------
// x0 = normalize(transpose(x)) into state layout [col][n], col = b*16 + ch.
// ---------------------------------------------------------------------------
__global__ void x0_kernel(const float* __restrict__ x,
                          float* __restrict__ X,
                          int N, int B, int C) {
    int idx = blockIdx.x * blockDim.x + threadIdx.x;
    int pairs = C / 2;
    int total = B * pairs * N;
    if (idx >= total) return;
    int n = idx % N;
    int t = idx / N;
    int p = t % pairs;
    int b = t / pairs;
    float a = x[((size_t)b * C + 2 * p) * N + n];
    float v = x[((size_t)b * C + 2 * p + 1) * N + n];
    float r = sqrtf(a * a + v * v) + 1e-6f;
    X[((size_t)(b * C + 2 * p)) * N + n]     = a / r;
    X[((size_t)(b * C + 2 * p + 1)) * N + n] = v / r;
}

// ---------------------------------------------------------------------------
// A = sc * conn_w  (elementwise, vectorized)
// ---------------------------------------------------------------------------
__global__ void amul_kernel(const float* __restrict__ sc,
                            const float* __restrict__ cw,
                            float* __restrict__ A, size_t total4) {
    size_t i = (size_t)blockIdx.x * blockDim.x + threadIdx.x;
    if (i >= total4) return;
    float4 a = ((const float4*)sc)[i];
    float4 b = ((const float4*)cw)[i];
    float4 r;
    r.x = a.x * b.x; r.y = a.y * b.y; r.z = a.z * b.z; r.w = a.w * b.w;
    ((float4*)A)[i] = r;
}

template <bool PRE>
__device__ __forceinline__ v2f loadA(const float* __restrict__ Apre,
                                     const float* __restrict__ sc,
                                     const float* __restrict__ cw,
                                     size_t off) {
    if (PRE) {
        return *(const v2f*)(Apre + off);
    } else {
        v2f s  = *(const v2f*)(sc + off);
        v2f c2 = *(const v2f*)(cw + off);
        return s * c2;
    }
}

// ---------------------------------------------------------------------------
// Split-K WMMA GEMM: partial coup = A[16 rows, Kslice] @ X[Kslice, 32].
// blockIdx.x = m-tile, blockIdx.y = K-slice.  One wave; both batches reuse
// the A registers.  4-stage software pipeline with explicitly named register
// sets: 12 loads in flight, each WMMA pair consumes operands loaded three
// stages earlier.
//
// WMMA f32 layouts (wave32):
//   A 16x4 : lane L holds A[m = L&15][k0], A[m][k0+1], k0 = kbase + 2*(L>>4)
//   B 4x16 : lane L holds B[k0][n = L&15], B[k0+1][n]
//   C/D    : vgpr j, lanes 0-15: M=j, N=lane; lanes 16-31: M=j+8, N=lane-16
// ---------------------------------------------------------------------------
template <bool PRE>
__global__ void gemm_kernel(const float* __restrict__ Apre,
                            const float* __restrict__ sc,
                            const float* __restrict__ cw,
                            const float* __restrict__ Xcur,
                            float* __restrict__ coupP,
                            int N) {
    const int lane  = threadIdx.x;
    const int row   = lane & 15;
    const int half  = lane >> 4;
    const int mtile = blockIdx.x;
    const int slc   = blockIdx.y;
    const int Ks    = N / SPLITK;            // multiple of 16
    const int k0    = slc * Ks;

    const size_t aOff  = ((size_t)(mtile * 16 + row)) * N + 2 * half + k0;
    const size_t b0Off = ((size_t)row) * N + 2 * half + k0;         // b=0 cols
    const size_t b1Off = ((size_t)(16 + row)) * N + 2 * half + k0;  // b=1 cols

    v8f acc0 = {0.f, 0.f, 0.f, 0.f, 0.f, 0.f, 0.f, 0.f};
    v8f acc1 = {0.f, 0.f, 0.f, 0.f, 0.f, 0.f, 0.f, 0.f};

#define LOAD3(av, x0v, x1v, kk)                                         \
    av  = loadA<PRE>(Apre, sc, cw, aOff + (kk));                        \
    x0v = *(const v2f*)(Xcur + b0Off + (kk));                           \
    x1v = *(const v2f*)(Xcur + b1Off + (kk));

#define CONSUME(av, x0v, x1v)                                           \
    acc0 = __builtin_amdgcn_wmma_f32_16x16x4_f32(false, av, false, x0v, \
                                                 (short)0, acc0, false, false); \
    acc1 = __builtin_amdgcn_wmma_f32_16x16x4_f32(false, av, false, x1v, \
                                                 (short)0, acc1, false, false);

    v2f a0, a1, a2, a3, p0, p1, p2, p3, q0, q1, q2, q3;
    LOAD3(a0, p0, q0, 0)
    LOAD3(a1, p1, q1, 4)
    LOAD3(a2, p2, q2, 8)
    LOAD3(a3, p3, q3, 12)

#pragma unroll 2
    for (int k = 16; k < Ks; k += 16) {
        CONSUME(a0, p0, q0)  LOAD3(a0, p0, q0, k)
        CONSUME(a1, p1, q1)  LOAD3(a1, p1, q1, k + 4)
        CONSUME(a2, p2, q2)  LOAD3(a2, p2, q2, k + 8)
        CONSUME(a3, p3, q3)  LOAD3(a3, p3, q3, k + 12)
    }
    CONSUME(a0, p0, q0)
    CONSUME(a1, p1, q1)
    CONSUME(a2, p2, q2)
    CONSUME(a3, p3, q3)

#undef LOAD3
#undef CONSUME

    // Store partial tile: lane holds col c = row, 8 consecutive m values.
    const int c     = row;
    const int mbase = mtile * 16 + 8 * half;
#pragma unroll
    for (int b = 0; b < 2; ++b) {
        v8f acc = b ? acc1 : acc0;
        size_t base = ((size_t)slc * 32 + b * 16 + c) * N + mbase;
        float4 lo = {acc[0], acc[1], acc[2], acc[3]};
        float4 hi = {acc[4], acc[5], acc[6], acc[7]};
        *(float4*)(coupP + base)     = lo;
        *(float4*)(coupP + base + 4) = hi;
    }
}

// ---------------------------------------------------------------------------
// Per-step update: sum split-K partials, tangent projection, omega rotation,
// pair-normalize.  One thread per (b, pair, n); both channels in-thread.
// ---------------------------------------------------------------------------
__global__ void update_kernel(const float* __restrict__ coupP,
                              const float* __restrict__ Xcur,
                              const float* __restrict__ y,
                              const float* __restrict__ omgp,
                              const float* __restrict__ gammap,
                              float* __restrict__ Xnext,
                              float* __restrict__ outq,
                              int N, int B, int C) {
    int idx = blockIdx.x * blockDim.x + threadIdx.x;
    int pairs = C / 2;
    int total = B * pairs * N;
    if (idx >= total) return;
    int n = idx % N;
    int t = idx / N;
    int p = t % pairs;
    int b = t / pairs;
    int c0 = 2 * p;

    size_t o0 = ((size_t)(b * C + c0)) * N + n;
    size_t o1 = o0 + N;
    size_t colN = (size_t)B * C * N;
    float coup0 = 0.f, coup1 = 0.f;
#pragma unroll
    for (int s = 0; s < SPLITK; ++s) {
        coup0 += coupP[(size_t)s * colN + o0];
        coup1 += coupP[(size_t)s * colN + o1];
    }
    float x0 = Xcur[o0], x1 = Xcur[o1];
    float yt0 = coup0 + y[o0];
    float yt1 = coup1 + y[o1];
    float sim = x0 * yt0 + x1 * yt1;          // <x, y> over the pair
    float pr0 = yt0 - sim * x0;               // tangent projection
    float pr1 = yt1 - sim * x1;
    float om = fabsf(omgp[p]);
    float g  = gammap[0];
    float u0 = x0 + g * ( om * x1 + pr0);     // omega rotation + step
    float u1 = x1 + g * (-om * x0 + pr1);
    float r = sqrtf(u0 * u0 + u1 * u1) + 1e-6f;
    float xn0 = u0 / r, xn1 = u1 / r;
    Xnext[o0] = xn0;
    Xnext[o1] = xn1;
    outq[((size_t)(b * N + n)) * C + c0]     = xn0;
    outq[((size_t)(b * N + n)) * C + c0 + 1] = xn1;
}

extern "C" void kernel_launch(void* const* d_in, const int* in_sizes, int n_in,
                              void* d_out, int out_size, void* d_ws, size_t ws_size,
                              hipStream_t stream) {
    const float* x    = (const float*)d_in[0];
    const float* cin  = (const float*)d_in[1];
    const float* sc   = (const float*)d_in[2];
    const float* gnw  = (const float*)d_in[3];
    const float* gnb  = (const float*)d_in[4];
    const float* cw   = (const float*)d_in[5];
    const float* omg  = (const float*)d_in[6];
    const float* gam  = (const float*)d_in[7];

    const int N = (int)(sqrt((double)in_sizes[2]) + 0.5);   // sc is [1,N,N]
    const int C = in_sizes[3];                               // 16
    const int B = in_sizes[0] / (C * N);                     // 2
    const int Q = out_size / (B * N * C);                    // 8

    const size_t colN = (size_t)B * C * N;                   // 32 * 8192
    float* y     = (float*)d_ws;
    float* X0    = y + colN;
    float* X1    = X0 + colN;
    float* coupP = X1 + colN;                                // SPLITK * colN
    float* Apre  = coupP + (size_t)SPLITK * colN;
    const bool pre =
        ws_size >= ((3 + SPLITK) * colN + (size_t)N * N) * sizeof(float);

    // 1) GroupNorm(c) -> y  (constant across steps)
    gn_kernel<<<B * (C / 2), 256, 0, stream>>>(cin, gnw, gnb, y, N, C / 2, C);

    // 2) x0 = normalize(x^T)
    {
        int total = B * (C / 2) * N;
        x0_kernel<<<(total + 255) / 256, 256, 0, stream>>>(x, X0, N, B, C);
    }

    // 3) optionally precompute A = sc * conn_w (halves per-step HBM traffic)
    if (pre) {
        size_t total4 = (size_t)N * N / 4;
        amul_kernel<<<(unsigned)((total4 + 255) / 256), 256, 0, stream>>>(sc, cw, Apre, total4);
    }

    // 4) Q steps: split-K WMMA GEMM + fused pointwise update
    dim3 ggrid(N / 16, SPLITK);
    int utotal = B * (C / 2) * N;
    int ublocks = (utotal + 255) / 256;
    for (int q = 0; q < Q; ++q) {
        const float* Xc = (q & 1) ? X1 : X0;
        float* Xn       = (q & 1) ? X0 : X1;
        float* outq = (float*)d_out + (size_t)q * B * N * C;
        if (pre)
            gemm_kernel<true><<<ggrid, 32, 0, stream>>>(Apre, sc, cw, Xc, coupP, N);
        else
            gemm_kernel<false><<<ggrid, 32, 0, stream>>>(nullptr, sc, cw, Xc, coupP, N);
        update_kernel<<<ublocks, 256, 0, stream>>>(coupP, Xc, y, omg, gam,
                                                   Xn, outq, N, B, C);
    }
}